// DeformableConv2d_guide_15951508537428
// MI455X (gfx1250) — compile-verified
//
#include <hip/hip_runtime.h>
#include <math.h>

typedef __attribute__((ext_vector_type(2))) float v2f;
typedef __attribute__((ext_vector_type(8))) float v8f;

#define B_   4
#define C_   64
#define H_   128
#define W_   128
#define O_   64
#define KK_  9
#define HW_  (H_ * W_)
#define K1_  (2 * C_ * KK_)   // 1152 : K of offset/mod conv
#define K2_  (C_ * KK_)       // 576  : K of deform einsum
#define MPAD_ 32              // 27 offset+mod channels padded to 32

// ---------------------------------------------------------------------------
// Pack w_off (18x1152) and w_mod (9x1152) into one 32x1152 matrix + bias[32].
// Both are already laid out with K index = ic*9 + ky*3 + kx (contiguous rows).
// ---------------------------------------------------------------------------
__global__ void pack_weights(const float* __restrict__ w_off,
                             const float* __restrict__ b_off,
                             const float* __restrict__ w_mod,
                             const float* __restrict__ b_mod,
                             float* __restrict__ Wp,
                             float* __restrict__ Bp) {
  int tid = blockIdx.x * blockDim.x + threadIdx.x;
  if (tid < MPAD_ * K1_) {
    int row = tid / K1_, k = tid % K1_;
    float v = 0.f;
    if (row < 18)      v = w_off[row * K1_ + k];
    else if (row < 27) v = w_mod[(row - 18) * K1_ + k];
    Wp[tid] = v;
  }
  if (tid < MPAD_) {
    float v = 0.f;
    if (tid < 18)      v = b_off[tid];
    else if (tid < 27) v = b_mod[tid - 18];
    Bp[tid] = v;
  }
}

// ---------------------------------------------------------------------------
// Kernel 1: offset + modulator 3x3 conv over concat(x, guide).
// WG = 128 threads (4 wave32), covers 32 pixels of one row x 32 channels.
// K=1152 processed in 8 chunks of 144 (16 input channels x 9 taps).
// Each wave owns one 16x16 tile: waves 0/1 -> M 0-15/16-31 @ N 0-15,
// waves 2/3 -> same M @ N 16-31. WMMA: V_WMMA_F32_16X16X4_F32.
// ---------------------------------------------------------------------------
constexpr int ICCH = 16;
constexpr int KCH1 = ICCH * KK_;  // 144
constexpr int NP1  = 32;

__global__ __launch_bounds__(128)
void offmod_conv(const float* __restrict__ x, const float* __restrict__ g,
                 const float* __restrict__ Wp, const float* __restrict__ Bp,
                 float* __restrict__ off_out, float* __restrict__ mod_out) {
  __shared__ float Alds[MPAD_][KCH1 + 1];  // +1 pad: row stride 145 -> no bank conflict
  __shared__ float Blds[KCH1][NP1 + 1];

  int blk = blockIdx.x;            // 2048 blocks
  int b   = blk >> 9;              // 512 blocks per batch (128 rows * 4 tiles)
  int rem = blk & 511;
  int y   = rem >> 2;
  int x0  = (rem & 3) * NP1;

  int tid  = threadIdx.x;
  int lane = tid & 31, wave = tid >> 5;
  int mtile = (wave & 1) * 16;
  int ntile = (wave >> 1) * 16;
  int mrow  = lane & 15;           // M (A,C/D) or N (B,C/D) index within tile
  int half  = lane >> 4;           // selects K pair / M+8 per ISA layout

  v8f acc = {};

  for (int ch = 0; ch < 8; ++ch) {
    int ic0 = ch * ICCH;
    // stage A: 32 x 144 weight chunk
    for (int e = tid; e < MPAD_ * KCH1; e += 128) {
      int m = e / KCH1, kr = e % KCH1;
      Alds[m][kr] = Wp[m * K1_ + ic0 * KK_ + kr];
    }
    // stage B: im2col 144 x 32 (zero-padded borders), fea = concat(x, guide)
    for (int e = tid; e < KCH1 * NP1; e += 128) {
      int kr = e / NP1, p = e % NP1;
      int ic = ic0 + kr / KK_;
      int r  = kr % KK_;
      int yy = y + (r / 3) - 1;
      int xx = x0 + p + (r % 3) - 1;
      float v = 0.f;
      if (yy >= 0 && yy < H_ && xx >= 0 && xx < W_) {
        const float* src = (ic < C_) ? (x + ((size_t)(b * C_ + ic)) * HW_)
                                     : (g + ((size_t)(b * C_ + (ic - C_))) * HW_);
        v = src[yy * W_ + xx];
      }
      Blds[kr][p] = v;
    }
    __syncthreads();
    // 36 WMMA steps per chunk; A: 16x4 f32 frag, B: 4x16 f32 frag
    for (int ks = 0; ks < KCH1; ks += 4) {
      v2f a, bb;
      a.x  = Alds[mtile + mrow][ks + 2 * half];
      a.y  = Alds[mtile + mrow][ks + 2 * half + 1];
      bb.x = Blds[ks + 2 * half][ntile + mrow];
      bb.y = Blds[ks + 2 * half + 1][ntile + mrow];
      acc = __builtin_amdgcn_wmma_f32_16x16x4_f32(
          false, a, false, bb, (short)0, acc, false, false);
    }
    __syncthreads();
  }

  // epilogue: C/D layout -> VGPR r holds M = r + 8*half, N = lane%16
  int n = x0 + ntile + mrow;
  for (int r = 0; r < 8; ++r) {
    int m = mtile + r + 8 * half;  // output channel (0..31, 27..31 discarded)
    float v = acc[r] + Bp[m];
    size_t pix = (size_t)y * W_ + n;
    if (m < 18) {
      off_out[((size_t)b * 18 + m) * HW_ + pix] = v;
    } else if (m < 27) {
      mod_out[((size_t)b * KK_ + (m - 18)) * HW_ + pix] =
          2.f / (1.f + __expf(-v));
    }
  }
}

// ---------------------------------------------------------------------------
// Kernel 2: fused deformable bilinear gather + einsum 'bckhw,ock->bohw'.
// WG = 128 threads (4 waves) covers 16 pixels; each wave computes one
// 16(out-ch) x 16(pixel) tile over K=576 in 4 chunks of 144 (16 ch x 9 taps).
// Bilinear coords/masks are pixel x tap only -> precomputed once in LDS.
// ---------------------------------------------------------------------------
constexpr int CCH  = 16;
constexpr int KCH2 = CCH * KK_;   // 144
constexpr int NP2  = 16;

__global__ __launch_bounds__(128)
void deform_wmma(const float* __restrict__ x, const float* __restrict__ w_reg,
                 const float* __restrict__ off_in, const float* __restrict__ mod_in,
                 float* __restrict__ out) {
  __shared__ float Alds[O_][KCH2 + 1];        // 64 x 145
  __shared__ float Vlds[KCH2][NP2 + 1];       // 144 x 17
  __shared__ float Coord[KK_ * NP2][3];       // {py, px, mask} per (tap, pixel)

  int blk = blockIdx.x;            // 4096 blocks
  int b   = blk >> 10;             // 1024 per batch (128 rows * 8 tiles)
  int rem = blk & 1023;
  int y   = rem >> 3;
  int x0  = (rem & 7) * NP2;

  int tid  = threadIdx.x;
  int lane = tid & 31, wave = tid >> 5;
  int mtile = wave * 16;
  int mrow  = lane & 15;
  int half  = lane >> 4;

  // Precompute sampling coordinates: py = off_y + ky + y - pad, etc.
  if (tid < KK_ * NP2) {
    int k = tid >> 4, p = tid & 15;
    int px_ = x0 + p;
    size_t pix = (size_t)y * W_ + px_;
    float oy = off_in[((size_t)b * 18 + 2 * k) * HW_ + pix];
    float ox = off_in[((size_t)b * 18 + 2 * k + 1) * HW_ + pix];
    float mk = mod_in[((size_t)b * KK_ + k) * HW_ + pix];
    Coord[tid][0] = oy + (float)(k / 3) + (float)y - 1.f;
    Coord[tid][1] = ox + (float)(k % 3) + (float)px_ - 1.f;
    Coord[tid][2] = mk;
  }
  __syncthreads();

  v8f acc = {};
  for (int ch = 0; ch < 4; ++ch) {
    int c0 = ch * CCH;
    // stage A: w_reg rows (already contiguous as [O][C*9])
    for (int e = tid; e < O_ * KCH2; e += 128) {
      int o = e / KCH2, kr = e % KCH2;
      Alds[o][kr] = w_reg[(size_t)o * K2_ + c0 * KK_ + kr];
    }
    // stage val: bilinear gather * mask, K row = c*9 + k
    for (int e = tid; e < KCH2 * NP2; e += 128) {
      int kr = e >> 4, p = e & 15;
      int c = c0 + kr / KK_;
      int k = kr % KK_;
      float py = Coord[k * NP2 + p][0];
      float px = Coord[k * NP2 + p][1];
      float mk = Coord[k * NP2 + p][2];
      float fy = floorf(py), fx = floorf(px);
      int y0 = (int)fy, x0i = (int)fx;
      float wy1 = py - fy, wx1 = px - fx;
      float wy0 = 1.f - wy1, wx0 = 1.f - wx1;
      const float* src = x + ((size_t)b * C_ + c) * HW_;
      bool yi0 = (y0 >= 0) & (y0 < H_);
      bool yi1 = (y0 + 1 >= 0) & (y0 + 1 < H_);
      bool xi0 = (x0i >= 0) & (x0i < W_);
      bool xi1 = (x0i + 1 >= 0) & (x0i + 1 < W_);
      float v00 = (yi0 & xi0) ? src[y0 * W_ + x0i] : 0.f;
      float v01 = (yi0 & xi1) ? src[y0 * W_ + x0i + 1] : 0.f;
      float v10 = (yi1 & xi0) ? src[(y0 + 1) * W_ + x0i] : 0.f;
      float v11 = (yi1 & xi1) ? src[(y0 + 1) * W_ + x0i + 1] : 0.f;
      float v = (v00 * wx0 + v01 * wx1) * wy0 + (v10 * wx0 + v11 * wx1) * wy1;
      Vlds[kr][p] = v * mk;
    }
    __syncthreads();
    for (int ks = 0; ks < KCH2; ks += 4) {
      v2f a, bb;
      a.x  = Alds[mtile + mrow][ks + 2 * half];
      a.y  = Alds[mtile + mrow][ks + 2 * half + 1];
      bb.x = Vlds[ks + 2 * half][mrow];
      bb.y = Vlds[ks + 2 * half + 1][mrow];
      acc = __builtin_amdgcn_wmma_f32_16x16x4_f32(
          false, a, false, bb, (short)0, acc, false, false);
    }
    __syncthreads();
  }

  for (int r = 0; r < 8; ++r) {
    int o = mtile + r + 8 * half;
    out[(((size_t)b * O_ + o) * H_ + y) * W_ + x0 + mrow] = acc[r];
  }
}

// ---------------------------------------------------------------------------
extern "C" void kernel_launch(void* const* d_in, const int* in_sizes, int n_in,
                              void* d_out, int out_size, void* d_ws, size_t ws_size,
                              hipStream_t stream) {
  const float* x     = (const float*)d_in[0];
  const float* guide = (const float*)d_in[1];
  const float* w_off = (const float*)d_in[2];
  const float* b_off = (const float*)d_in[3];
  const float* w_mod = (const float*)d_in[4];
  const float* b_mod = (const float*)d_in[5];
  const float* w_reg = (const float*)d_in[6];
  float* out = (float*)d_out;

  // workspace layout (floats): packed weights | bias | offset | modulator
  float* ws     = (float*)d_ws;
  float* Wp     = ws;                              // 32*1152
  float* Bp     = Wp + (size_t)MPAD_ * K1_;        // 32
  float* off_ws = Bp + MPAD_;                      // 4*18*16384
  float* mod_ws = off_ws + (size_t)B_ * 18 * HW_;  // 4*9*16384   (~7.3 MB total)

  pack_weights<<<(MPAD_ * K1_ + 255) / 256, 256, 0, stream>>>(
      w_off, b_off, w_mod, b_mod, Wp, Bp);
  offmod_conv<<<(B_ * HW_) / NP1, 128, 0, stream>>>(
      x, guide, Wp, Bp, off_ws, mod_ws);
  deform_wmma<<<(B_ * HW_) / NP2, 128, 0, stream>>>(
      x, w_reg, off_ws, mod_ws, out);
}